// PureMambaBlock_71717363908667
// MI455X (gfx1250) — compile-verified
//
#include <hip/hip_runtime.h>
#include <hip/hip_bf16.h>
#include <math.h>

typedef __attribute__((ext_vector_type(2))) float v2f;
typedef __attribute__((ext_vector_type(8))) float v8f;

#define BATCH   2
#define SEQLEN  1024
#define D_MODEL 512
#define D_INNER 1024
#define D_STATE 128
#define DT_RANK 32
#define NPROJ   (DT_RANK + 2 * D_STATE)   // 288
#define ROWS    (BATCH * SEQLEN)          // 2048

__device__ __forceinline__ float silu_f(float x) {
    return x * (1.0f / (1.0f + __expf(-x)));
}
__device__ __forceinline__ float softplus_f(float x) {
    return (x > 20.0f) ? x : log1pf(__expf(x));
}

// ---------------------------------------------------------------------------
// f32 WMMA GEMM: C[M,N] = A[M,K] @ B[K,N] (row major).
// Each wave32 computes a 16 x (NB*16) strip: the A fragment is loaded once
// per k-step and reused across NB back-to-back V_WMMA_F32_16X16X4_F32 ops,
// cutting A-side traffic NB-fold and raising wmma density per vmem op.
// EPI: 0 = plain store, 1 = silu on cols < split, 2 = softplus(v + bias[col]).
// ---------------------------------------------------------------------------
template <int EPI, int NB>
__global__ __launch_bounds__(32) void gemm16_wmma_f32(
    const float* __restrict__ A, int lda,
    const float* __restrict__ B, int ldb,
    float* __restrict__ C, int ldc,
    int K, const float* __restrict__ bias, int split)
{
    const int tn   = blockIdx.x;        // strip of NB N-tiles
    const int tm   = blockIdx.y;        // M tile index
    const int lane = threadIdx.x;       // 0..31
    const int half = lane >> 4;         // 0: K=0,1   1: K=2,3
    const int l16  = lane & 15;

    // A: lane l16 owns row (tm*16+l16); a.x/a.y = K = 2*half, 2*half+1
    // B: lane l16 owns col (tile*16+l16); b.x/b.y = K = 2*half, 2*half+1
    const float* Arow = A + (long)(tm * 16 + l16) * lda;
    const float* Bcol = B + (tn * (NB * 16) + l16);

    v8f acc[NB];
#pragma unroll
    for (int nb = 0; nb < NB; ++nb) acc[nb] = (v8f){};

    for (int k0 = 0; k0 < K; k0 += 4) {
        v2f a;
        a.x = Arow[k0 + 2 * half];
        a.y = Arow[k0 + 2 * half + 1];
        const float* Bk0 = Bcol + (long)(k0 + 2 * half) * ldb;
        const float* Bk1 = Bcol + (long)(k0 + 2 * half + 1) * ldb;
#pragma unroll
        for (int nb = 0; nb < NB; ++nb) {
            v2f b;
            b.x = Bk0[nb * 16];
            b.y = Bk1[nb * 16];
            acc[nb] = __builtin_amdgcn_wmma_f32_16x16x4_f32(
                /*neg_a=*/false, a, /*neg_b=*/false, b,
                /*c_mod=*/(short)0, acc[nb], /*reuse_a=*/false, /*reuse_b=*/false);
        }
    }

    // D layout: lanes 0-15 -> rows 0..7 in VGPR 0..7; lanes 16-31 -> rows 8..15
#pragma unroll
    for (int nb = 0; nb < NB; ++nb) {
        const int col = tn * (NB * 16) + nb * 16 + l16;
#pragma unroll
        for (int r = 0; r < 8; ++r) {
            const int row = tm * 16 + half * 8 + r;
            float v = acc[nb][r];
            if (EPI == 1) {
                if (col < split) v = silu_f(v);
            } else if (EPI == 2) {
                v = softplus_f(v + bias[col]);
            }
            C[(long)row * ldc + col] = v;
        }
    }
}

// ---------------------------------------------------------------------------
// Selective scan: one wave32 per (batch, d_inner-channel). Lane owns the
// contiguous states n = 4*lane .. 4*lane+3, so B and C are each a single
// aligned float4 (global_load_b128) per step. 1024 dependent steps; all
// per-step operands are L2-resident (192 MB >> 40 MB working set).
// ---------------------------------------------------------------------------
__global__ __launch_bounds__(32) void mamba_scan(
    const float* __restrict__ dt, const float* __restrict__ proj,
    const float* __restrict__ UZ, const float* __restrict__ A_log,
    const float* __restrict__ D_param, float* __restrict__ y)
{
    const int bd   = blockIdx.x;             // 0 .. BATCH*D_INNER-1
    const int b    = bd >> 10;
    const int d    = bd & (D_INNER - 1);
    const int lane = threadIdx.x;

    const float4 al = *(const float4*)(A_log + d * D_STATE + lane * 4);
    float a[4], h[4];
    a[0] = -__expf(al.x); a[1] = -__expf(al.y);
    a[2] = -__expf(al.z); a[3] = -__expf(al.w);
#pragma unroll
    for (int j = 0; j < 4; ++j) h[j] = 0.0f;
    const float Dp = D_param[d];

    for (int t = 0; t < SEQLEN; ++t) {
        const int   row = b * SEQLEN + t;
        const float dtv = dt[(long)row * D_INNER + d];
        const float uv  = UZ[(long)row * (2 * D_INNER) + d];
        const float* pr = proj + (long)row * NPROJ;
        if (t + 1 < SEQLEN) {
            __builtin_prefetch(pr + NPROJ, 0, 1);   // next step's B/C row
        }
        const float4 Bv = *(const float4*)(pr + DT_RANK + lane * 4);
        const float4 Cv = *(const float4*)(pr + DT_RANK + D_STATE + lane * 4);
        const float du = dtv * uv;
        float acc;
        h[0] = __expf(dtv * a[0]) * h[0] + du * Bv.x;
        h[1] = __expf(dtv * a[1]) * h[1] + du * Bv.y;
        h[2] = __expf(dtv * a[2]) * h[2] + du * Bv.z;
        h[3] = __expf(dtv * a[3]) * h[3] + du * Bv.w;
        acc = h[0] * Cv.x + h[1] * Cv.y + h[2] * Cv.z + h[3] * Cv.w;
#pragma unroll
        for (int off = 16; off; off >>= 1) acc += __shfl_xor(acc, off, 32);
        if (lane == 0) y[(long)row * D_INNER + d] = acc + Dp * uv;
    }
}

// ---------------------------------------------------------------------------
// LayerNorm over D_INNER + silu(z) gate, in place on y. Contiguous float4
// per thread (256 threads x 4 = 1024 columns).
// ---------------------------------------------------------------------------
__global__ __launch_bounds__(256) void ln_gate(
    float* __restrict__ y, const float* __restrict__ UZ,
    const float* __restrict__ gamma, const float* __restrict__ beta)
{
    const int row = blockIdx.x;
    const int tid = threadIdx.x;
    const int c0  = tid * 4;

    const float4 v = *(const float4*)(y + (long)row * D_INNER + c0);
    float s  = v.x + v.y + v.z + v.w;
    float s2 = v.x * v.x + v.y * v.y + v.z * v.z + v.w * v.w;

#pragma unroll
    for (int off = 16; off; off >>= 1) {
        s  += __shfl_xor(s, off, 32);
        s2 += __shfl_xor(s2, off, 32);
    }
    __shared__ float rs[8], rs2[8];
    const int wid = tid >> 5;
    if ((tid & 31) == 0) { rs[wid] = s; rs2[wid] = s2; }
    __syncthreads();
    s = 0.0f; s2 = 0.0f;
#pragma unroll
    for (int w = 0; w < 8; ++w) { s += rs[w]; s2 += rs2[w]; }

    const float mu   = s * (1.0f / D_INNER);
    const float var  = s2 * (1.0f / D_INNER) - mu * mu;
    const float rstd = rsqrtf(var + 1e-5f);

    const float4 ga = *(const float4*)(gamma + c0);
    const float4 be = *(const float4*)(beta + c0);
    const float4 zv = *(const float4*)(UZ + (long)row * (2 * D_INNER) + D_INNER + c0);

    float4 o;
    o.x = ((v.x - mu) * rstd * ga.x + be.x) * silu_f(zv.x);
    o.y = ((v.y - mu) * rstd * ga.y + be.y) * silu_f(zv.y);
    o.z = ((v.z - mu) * rstd * ga.z + be.z) * silu_f(zv.z);
    o.w = ((v.w - mu) * rstd * ga.w + be.w) * silu_f(zv.w);
    *(float4*)(y + (long)row * D_INNER + c0) = o;
}

extern "C" void kernel_launch(void* const* d_in, const int* in_sizes, int n_in,
                              void* d_out, int out_size, void* d_ws, size_t ws_size,
                              hipStream_t stream)
{
    const float* x     = (const float*)d_in[0];
    const float* W_in  = (const float*)d_in[1];
    const float* W_x   = (const float*)d_in[2];
    const float* W_dt  = (const float*)d_in[3];
    const float* b_dt  = (const float*)d_in[4];
    const float* W_out = (const float*)d_in[5];
    const float* gamma = (const float*)d_in[6];
    const float* beta  = (const float*)d_in[7];
    const float* A_log = (const float*)d_in[8];
    const float* D_par = (const float*)d_in[9];
    float* out = (float*)d_out;

    float* UZ   = (float*)d_ws;                       // ROWS x 2*D_INNER
    float* proj = UZ   + (size_t)ROWS * 2 * D_INNER;  // ROWS x NPROJ
    float* dtb  = proj + (size_t)ROWS * NPROJ;        // ROWS x D_INNER
    float* yb   = dtb  + (size_t)ROWS * D_INNER;      // ROWS x D_INNER

    // 1. UZ = x @ W_in ; silu on the u half (cols < D_INNER). N=2048, NB=4.
    gemm16_wmma_f32<1, 4><<<dim3((2 * D_INNER) / 64, ROWS / 16), 32, 0, stream>>>(
        x, D_MODEL, W_in, 2 * D_INNER, UZ, 2 * D_INNER, D_MODEL, nullptr, D_INNER);

    // 2. proj = u @ W_x   (u = UZ with row stride 2*D_INNER). N=288=9*32, NB=2.
    gemm16_wmma_f32<0, 2><<<dim3(NPROJ / 32, ROWS / 16), 32, 0, stream>>>(
        UZ, 2 * D_INNER, W_x, NPROJ, proj, NPROJ, D_INNER, nullptr, 0);

    // 3. dt = softplus(proj[:, :DT_RANK] @ W_dt + b_dt). N=1024, NB=4.
    gemm16_wmma_f32<2, 4><<<dim3(D_INNER / 64, ROWS / 16), 32, 0, stream>>>(
        proj, NPROJ, W_dt, D_INNER, dtb, D_INNER, DT_RANK, b_dt, 0);

    // 4. selective scan: one wave per (b, d)
    mamba_scan<<<BATCH * D_INNER, 32, 0, stream>>>(dtb, proj, UZ, A_log, D_par, yb);

    // 5. LayerNorm + silu(z) gate, in place on yb
    ln_gate<<<ROWS, 256, 0, stream>>>(yb, UZ, gamma, beta);

    // 6. out = G @ W_out. N=512, NB=4.
    gemm16_wmma_f32<0, 4><<<dim3(D_MODEL / 64, ROWS / 16), 32, 0, stream>>>(
        yb, D_INNER, W_out, D_MODEL, out, D_MODEL, D_INNER, nullptr, 0);
}